// IoUInstanceLoss_46179488367044
// MI455X (gfx1250) — compile-verified
//
#include <hip/hip_runtime.h>
#include <hip/hip_bf16.h>

#define NI   16      // instances
#define MAXB 512     // max blocks feeding the partial matrix
#define BS   256     // block size for the main pass (8 wave32)

typedef float v2f __attribute__((ext_vector_type(2)));
typedef float v8f __attribute__((ext_vector_type(8)));

struct Ws {
  unsigned long long amax[NI];      // packed argmax keys: (center_bits<<32)|(~idx)
  int    counts[NI + 1];            // counts[0] = background count
  int    pad0;
  float2 pair[NI][32];              // {mean_j, -0.5*log2(e)/var_j}
  float  inv_sumvar[NI];
  float  countf[NI];
  float  nbg;
  float  pad1;
  float  partial[3][NI][MAXB];      // q=0: S_all, q=1: S_sel, q=2: pos count
};

__global__ void k_zero(Ws* ws) {
  int t = threadIdx.x;
  if (t < NI)     ws->amax[t]   = 0ull;
  if (t < NI + 1) ws->counts[t] = 0;
}

// Pass 1: per-instance count + argmax(centers under mask), block pre-reduced.
__global__ void k_stats(const float* __restrict__ centers,
                        const int*   __restrict__ labels,
                        Ws* ws, int n) {
  __shared__ unsigned long long samax[NI];
  __shared__ int scnt[NI + 1];
  int t = threadIdx.x;
  if (t < NI)     samax[t] = 0ull;
  if (t < NI + 1) scnt[t]  = 0;
  __syncthreads();
  int i = blockIdx.x * blockDim.x + t;
  if (i < n) {
    int l = labels[i];
    if ((unsigned)l <= (unsigned)NI) {
      atomicAdd(&scnt[l], 1);
      if (l >= 1) {
        // centers in [0,1): non-negative float bits order like ints.
        unsigned int cb = __float_as_uint(centers[i]);
        unsigned long long key =
            ((unsigned long long)cb << 32) | (unsigned long long)(0xFFFFFFFFu - (unsigned)i);
        atomicMax(&samax[l - 1], key);   // ties -> smallest i (first occurrence)
      }
    }
  }
  __syncthreads();
  if (t < NI && samax[t]) atomicMax(&ws->amax[t], samax[t]);
  if (t < NI + 1 && scnt[t]) atomicAdd(&ws->counts[t], scnt[t]);
}

// Pass 2: gather mean/var at argmax index; one wave32 per instance (lane = dim).
__global__ void k_prep(const float* __restrict__ emb29,
                       const float* __restrict__ pts,
                       const float* __restrict__ vars,
                       Ws* ws) {
  int tid = threadIdx.x;            // 512 threads = 16 waves
  int t = tid >> 5, j = tid & 31;
  int cnt = ws->counts[t + 1];
  unsigned long long key = ws->amax[t];
  int idx = (cnt > 0) ? (int)(0xFFFFFFFFu - (unsigned int)(key & 0xFFFFFFFFull)) : 0;
  float e = (j < 29) ? emb29[idx * 29 + j] : pts[idx * 3 + (j - 29)];
  float v = vars[idx * 32 + j] + 1e-8f;
  float sv = v;
  #pragma unroll
  for (int m = 16; m >= 1; m >>= 1) sv += __shfl_xor(sv, m);
  ws->pair[t][j] = make_float2(e, -0.72134752044448170367f / v); // -0.5*log2(e)/var
  if (j == 0) {
    ws->inv_sumvar[t] = 1.0f / sv;
    ws->countf[t]     = (float)cnt;
  }
  if (tid == 0) ws->nbg = (float)ws->counts[0];
}

// Pass 3 (hot): each thread keeps its 32-dim embedding in registers and sweeps
// all 16 instances; 48 f32 accumulators, wave-shuffle + LDS block reduction.
__global__ void __launch_bounds__(BS) k_main(const float* __restrict__ emb29,
                                             const float* __restrict__ pts,
                                             const int*   __restrict__ labels,
                                             Ws* ws, int n) {
  __shared__ float2 spair[NI * 32];            // 4 KB broadcast table
  __shared__ float  sinv[NI];
  __shared__ float  red[BS / 32][NI][3];
  int tid = threadIdx.x;
  if (tid < NI * 32) spair[tid] = ws->pair[tid >> 5][tid & 31];
  if (tid < NI)      sinv[tid]  = ws->inv_sumvar[tid];
  __syncthreads();

  float sAll[NI], sSel[NI], pCnt[NI];
  #pragma unroll
  for (int t = 0; t < NI; ++t) { sAll[t] = 0.f; sSel[t] = 0.f; pCnt[t] = 0.f; }

  for (int i = blockIdx.x * BS + tid; i < n; i += gridDim.x * BS) {
    float e[32];
    #pragma unroll
    for (int j = 0; j < 29; ++j) e[j] = emb29[i * 29 + j];
    #pragma unroll
    for (int j = 0; j < 3; ++j)  e[29 + j] = pts[i * 3 + j];
    int lab = labels[i];
    #pragma unroll
    for (int t = 0; t < NI; ++t) {
      float acc = 0.f;
      #pragma unroll
      for (int j = 0; j < 32; ++j) {
        float2 mv = spair[t * 32 + j];         // ds_load_b64 broadcast
        float d = e[j] - mv.x;
        acc += __builtin_amdgcn_exp2f(d * d * mv.y);   // v_exp_f32
      }
      float p   = acc * sinv[t];
      bool  isl = (lab == t + 1);
      float lb  = isl ? 1.f : 0.f;
      float df  = p - lb;
      float sq  = df * df;
      bool  pos = p > 0.5f;
      sAll[t] += sq;
      if (pos || isl) sSel[t] += sq;
      if (pos)        pCnt[t] += 1.f;
    }
  }

  int lane = tid & 31, w = tid >> 5;
  #pragma unroll
  for (int t = 0; t < NI; ++t) {
    float a = sAll[t], s = sSel[t], c = pCnt[t];
    #pragma unroll
    for (int m = 16; m >= 1; m >>= 1) {
      a += __shfl_xor(a, m); s += __shfl_xor(s, m); c += __shfl_xor(c, m);
    }
    if (lane == 0) { red[w][t][0] = a; red[w][t][1] = s; red[w][t][2] = c; }
  }
  __syncthreads();
  if (tid < NI) {
    float a = 0.f, s = 0.f, c = 0.f;
    #pragma unroll
    for (int ww = 0; ww < BS / 32; ++ww) {
      a += red[ww][tid][0]; s += red[ww][tid][1]; c += red[ww][tid][2];
    }
    ws->partial[0][tid][blockIdx.x] = a;
    ws->partial[1][tid][blockIdx.x] = s;
    ws->partial[2][tid][blockIdx.x] = c;
  }
}

// Pass 4: one wave. WMMA f32 16x16x4 reduces the 16xNB partial matrices with a
// ones B-matrix (exact f32), then lane 0 combines per-instance losses.
__global__ void k_final(const Ws* __restrict__ ws, float* __restrict__ out,
                        int n, int nblocks) {
  __shared__ float ssum[3][NI];
  int lane = threadIdx.x;                 // 32 threads, EXEC all-ones for WMMA
  int m  = lane & 15;                     // A-matrix row (instance)
  int kb = (lane >> 4) << 1;              // lanes 0-15 -> K=0,1 ; 16-31 -> K=2,3
  v2f bones; bones.x = 1.0f; bones.y = 1.0f;
  v8f cq[3];
  for (int q = 0; q < 3; ++q) {
    v8f c = {0.f, 0.f, 0.f, 0.f, 0.f, 0.f, 0.f, 0.f};
    for (int blk = 0; blk < nblocks; blk += 4) {
      int b0 = blk + kb, b1 = b0 + 1;
      v2f a;
      a.x = (b0 < nblocks) ? ws->partial[q][m][b0] : 0.f;
      a.y = (b1 < nblocks) ? ws->partial[q][m][b1] : 0.f;
      // D[m,n] = sum_k A[m,k] * 1 + C  -> accumulates block-dim reduction
      c = __builtin_amdgcn_wmma_f32_16x16x4_f32(false, a, false, bones,
                                                (short)0, c, false, false);
    }
    cq[q] = c;
  }
  // C/D layout: VGPR r, lanes 0-15 -> M=r (N=lane); lanes 16-31 -> M=r+8.
  // Column N=0 lives in lanes 0 and 16.
  #pragma unroll
  for (int q = 0; q < 3; ++q) {
    if (lane == 0) {
      #pragma unroll
      for (int r = 0; r < 8; ++r) ssum[q][r] = cq[q][r];
    }
    if (lane == 16) {
      #pragma unroll
      for (int r = 0; r < 8; ++r) ssum[q][8 + r] = cq[q][r];
    }
  }
  __syncthreads();
  if (lane == 0) {
    float nbg  = ws->nbg;
    float invn = 1.0f / (float)n;
    float total = 0.f;
    for (int t = 0; t < NI; ++t) {
      float cnt   = ws->countf[t];
      float P     = ssum[2][t];
      float ratio = nbg / (cnt + P);
      float loss  = (ssum[0][t] + ratio * ssum[1][t]) * invn;
      if (cnt > 0.f) total += loss;
    }
    out[0] = total;
  }
}

extern "C" void kernel_launch(void* const* d_in, const int* in_sizes, int n_in,
                              void* d_out, int out_size, void* d_ws, size_t ws_size,
                              hipStream_t stream) {
  const float* centers = (const float*)d_in[0];   // (N,1)
  const float* emb29   = (const float*)d_in[1];   // (N,29)
  const float* vars    = (const float*)d_in[2];   // (N,32)
  const float* pts     = (const float*)d_in[3];   // (N,3)
  const int*   labels  = (const int*)  d_in[4];   // (N,1)
  float* out = (float*)d_out;
  int n = in_sizes[0];
  Ws* ws = (Ws*)d_ws;

  int nb_stats = (n + BS - 1) / BS;
  int nb_main  = nb_stats > MAXB ? MAXB : nb_stats;

  k_zero <<<1, 64, 0, stream>>>(ws);
  k_stats<<<nb_stats, BS, 0, stream>>>(centers, labels, ws, n);
  k_prep <<<1, NI * 32, 0, stream>>>(emb29, pts, vars, ws);
  k_main <<<nb_main, BS, 0, stream>>>(emb29, pts, labels, ws, n);
  k_final<<<1, 32, 0, stream>>>(ws, out, n, nb_main);
}